// KNN_InnerProduct_1434519076948
// MI455X (gfx1250) — compile-verified
//
#include <hip/hip_runtime.h>
#include <stdint.h>

typedef __attribute__((ext_vector_type(2))) float v2f;
typedef __attribute__((ext_vector_type(8))) float v8f;
typedef __attribute__((ext_vector_type(4))) int i128_t;   // 128-bit payload type

#define AS1 __attribute__((address_space(1)))
#define AS3 __attribute__((address_space(3)))

#define KTOP 100
#define DIM 64
#define NCHUNK 64
#define RPS 128          // rows per stage
#define CPAD 68          // padded row stride (floats): 272B = 17*16B (b128-aligned, conflict-free)
#define CAP 16           // candidate buffer per query per round

#if defined(__HIP_DEVICE_COMPILE__) && __has_builtin(__builtin_amdgcn_global_load_async_to_lds_b128)
#define HAVE_ASYNC 1
#else
#define HAVE_ASYNC 0
#endif

__device__ __forceinline__ void wait_async(int n) {
#if HAVE_ASYNC
#if __has_builtin(__builtin_amdgcn_s_wait_asynccnt)
    if (n == 0) __builtin_amdgcn_s_wait_asynccnt(0);
    else        __builtin_amdgcn_s_wait_asynccnt(8);
#else
    if (n == 0) asm volatile("s_wait_asynccnt 0x0" ::: "memory");
    else        asm volatile("s_wait_asynccnt 0x8" ::: "memory");
#endif
#else
    (void)n;
#endif
}

__device__ __forceinline__ unsigned sortKey(float f) {
    unsigned b = __float_as_uint(f);
    return (b & 0x80000000u) ? ~b : (b | 0x80000000u);
}
__device__ __forceinline__ float unsortKey(unsigned s) {
    unsigned b = (s & 0x80000000u) ? (s ^ 0x80000000u) : ~s;
    return __uint_as_float(b);
}

// ---------------- Pass 1: WMMA scoring + per-chunk top-K ----------------
__global__ __launch_bounds__(256) void knn_score_topk(
    const float* __restrict__ qe, const float* __restrict__ corpus,
    unsigned long long* __restrict__ ws, int N, int CH)
{
    __shared__ float qtile[16 * CPAD];
    __shared__ float cstage[2][RPS * CPAD];   // double buffer for async pipeline
    __shared__ float sc[16 * RPS];
    __shared__ float topv[16][KTOP];
    __shared__ unsigned topi[16][KTOP];
    __shared__ float candv[16][CAP];
    __shared__ unsigned candi[16][CAP];
    __shared__ int ccnt[16];
    __shared__ float minv[16];

    const int tid  = threadIdx.x;
    const int lane = tid & 31;
    const int wv   = tid >> 5;       // wave 0..7 -> corpus sub-rows
    const int n16  = lane & 15;
    const int hi   = lane >> 4;      // 0/1 (lane half)
    const int qt   = blockIdx.x;     // query tile (16 queries)
    const int ck   = blockIdx.y;     // corpus chunk
    const int rowBeg = ck * CH;
    const int rowEnd = min(rowBeg + CH, N);

    // ---- stage-issue helper: 128 rows -> cstage[buf] (8 x b128 per thread) ----
    auto issue_stage = [&](int stageIdx, int buf) {
        const int rowBase = rowBeg + stageIdx * RPS;
        for (int f = tid; f < RPS * 16; f += 256) {
            int r = f >> 4, c4 = (f & 15) * 4;
            int g = rowBase + r;
            if (g > rowEnd - 1) g = rowEnd - 1;     // clamp: keep EXEC full, 8 uniform issues
            const float* src = corpus + (size_t)g * DIM + c4;
            float* dst = &cstage[buf][r * CPAD + c4];
#if HAVE_ASYNC
            __builtin_amdgcn_global_load_async_to_lds_b128(
                (AS1 i128_t*)src, (AS3 i128_t*)dst, 0, 0);
#else
            *(float4*)dst = *(const float4*)src;
#endif
        }
    };

    // stage query tile (coalesced float4, padded rows)
    {
        int r = tid >> 4, c4 = (tid & 15) * 4;
        float4 v = *(const float4*)(qe + (size_t)(qt * 16 + r) * DIM + c4);
        *(float4*)&qtile[r * CPAD + c4] = v;
    }
    if (tid < 16) ccnt[tid] = 0;
    issue_stage(0, 0);                    // prologue: async-fill buffer 0
    __syncthreads();

    // A fragments: 16x4 fp32 tiles. lane<16: K=k0,k0+1 ; lane>=16: K=k0+2,k0+3
    v2f afrag[16];
#pragma unroll
    for (int ks = 0; ks < 16; ++ks)
        afrag[ks] = *(const v2f*)&qtile[n16 * CPAD + ks * 4 + 2 * hi];

    int minpos = 0;                 // used by threads 0..15 (query = tid)
    const int myq = tid >> 4;       // phase-A: 16 threads per query
    const int sub = tid & 15;

    const int nStages = (CH + RPS - 1) / RPS;
    for (int it = 0; it < nStages; ++it) {
        const int rowBase = rowBeg + it * RPS;
        const int buf = it & 1;

        // pipeline: kick off next stage into other buffer, then wait for this stage
        if (it + 1 < nStages) {
            issue_stage(it + 1, buf ^ 1);
            wait_async(8);
        } else {
            wait_async(0);
        }
        __syncthreads();

        // 16x16 score tile per wave via fp32 WMMA (K stepped by 4)
        v8f acc = {0.f, 0.f, 0.f, 0.f, 0.f, 0.f, 0.f, 0.f};
        const float* bbase = &cstage[buf][(wv * 16 + n16) * CPAD + 2 * hi];
#pragma unroll
        for (int ks = 0; ks < 16; ++ks) {
            v2f b = *(const v2f*)(bbase + ks * 4);
            acc = __builtin_amdgcn_wmma_f32_16x16x4_f32(
                false, afrag[ks], false, b, (short)0, acc, false, false);
        }
        // C layout: lane<16 -> M=v ; lane>=16 -> M=v+8 ; N = lane%16
#pragma unroll
        for (int v = 0; v < 8; ++v)
            sc[(v + 8 * hi) * RPS + wv * 16 + n16] = acc[v];
        __syncthreads();

        if (it == 0) {              // warm start: seed top-100 with first rows
            if (tid < 16) {
                float mn = 3.4e38f; int mp = 0;
                for (int s = 0; s < KTOP; ++s) {
                    int g = rowBase + s;
                    float v = (g < rowEnd) ? sc[tid * RPS + s] : -3.4e38f;
                    topv[tid][s] = v;
                    topi[tid][s] = (g < rowEnd) ? (unsigned)g : 0u;
                    if (v < mn) { mn = v; mp = s; }
                }
                minv[tid] = mn; minpos = mp;
            }
            __syncthreads();
        }

        // threshold-filtered candidate collection + serial insertion
        const int jbeg = (it == 0) ? KTOP : 0;
        bool pend[8];
#pragma unroll
        for (int i = 0; i < 8; ++i) {
            int j = sub * 8 + i;
            pend[i] = (j >= jbeg) && (rowBase + j < rowEnd);
        }
        bool anyp = pend[0] | pend[1] | pend[2] | pend[3] |
                    pend[4] | pend[5] | pend[6] | pend[7];
        while (__syncthreads_or((int)anyp)) {
            // phase A: append candidates above threshold (LDS atomics)
#pragma unroll
            for (int i = 0; i < 8; ++i) {
                if (pend[i]) {
                    int j = sub * 8 + i;
                    float v = sc[myq * RPS + j];
                    if (v > minv[myq]) {
                        int pos = atomicAdd(&ccnt[myq], 1);
                        if (pos < CAP) {
                            candv[myq][pos] = v;
                            candi[myq][pos] = (unsigned)(rowBase + j);
                            pend[i] = false;
                        }
                    } else pend[i] = false;
                }
            }
            __syncthreads();
            // phase B: one thread per query inserts serially
            if (tid < 16) {
                int c = ccnt[tid]; if (c > CAP) c = CAP;
                float mn = minv[tid];
                for (int k = 0; k < c; ++k) {
                    float v = candv[tid][k];
                    if (v > mn) {
                        topv[tid][minpos] = v;
                        topi[tid][minpos] = candi[tid][k];
                        mn = topv[tid][0]; minpos = 0;
                        for (int s = 1; s < KTOP; ++s)
                            if (topv[tid][s] < mn) { mn = topv[tid][s]; minpos = s; }
                    }
                }
                minv[tid] = mn; ccnt[tid] = 0;
            }
            anyp = pend[0] | pend[1] | pend[2] | pend[3] |
                   pend[4] | pend[5] | pend[6] | pend[7];
        }
    }
    __syncthreads();

    // emit packed (sortable-score<<32 | row) candidates
    for (int e = tid; e < 16 * KTOP; e += 256) {
        int q = e / KTOP, s = e % KTOP;
        unsigned long long key =
            ((unsigned long long)sortKey(topv[q][s]) << 32) | topi[q][s];
        ws[(((size_t)(qt * 16 + q)) * NCHUNK + ck) * KTOP + s] = key;
    }
}

// ---------------- Pass 2: merge 6400 candidates -> sorted top-100 ----------------
__global__ __launch_bounds__(256) void knn_merge(
    const unsigned long long* __restrict__ ws,
    const float* __restrict__ corpus,
    const long long* __restrict__ cid,
    void* __restrict__ outBase, int B)
{
    __shared__ unsigned long long keys[8192];
    const int tid = threadIdx.x;
    const int q = blockIdx.x;
    const int NC = NCHUNK * KTOP;   // 6400
    for (int i = tid; i < 8192; i += 256)
        keys[i] = (i < NC) ? ws[(size_t)q * NC + i] : 0ull;
    __syncthreads();

    // descending bitonic sort of 8192 u64 keys
    for (int k = 2; k <= 8192; k <<= 1)
        for (int j = k >> 1; j > 0; j >>= 1) {
            for (int i = tid; i < 8192; i += 256) {
                int ixj = i ^ j;
                if (ixj > i) {
                    unsigned long long a = keys[i], b = keys[ixj];
                    bool up = ((i & k) == 0);
                    if (up ? (a < b) : (a > b)) { keys[i] = b; keys[ixj] = a; }
                }
            }
            __syncthreads();
        }

    long long* out_ids = (long long*)outBase;
    float* out_sc  = (float*)((char*)outBase + (size_t)B * KTOP * 8);
    float* out_emb = out_sc + (size_t)B * KTOP;

    for (int s = tid; s < KTOP; s += 256) {
        unsigned long long key = keys[s];
        unsigned row = (unsigned)key;
        out_ids[(size_t)q * KTOP + s] = cid[row];
        out_sc[(size_t)q * KTOP + s] = unsortKey((unsigned)(key >> 32));
    }
    for (int f = tid; f < KTOP * DIM; f += 256) {
        int s = f >> 6, c = f & 63;
        unsigned row = (unsigned)keys[s];
        out_emb[((size_t)q * KTOP + s) * DIM + c] = corpus[(size_t)row * DIM + c];
    }
}

extern "C" void kernel_launch(void* const* d_in, const int* in_sizes, int n_in,
                              void* d_out, int out_size, void* d_ws, size_t ws_size,
                              hipStream_t stream) {
    (void)n_in; (void)out_size; (void)ws_size;
    const float* qe = (const float*)d_in[0];
    const float* corpus = (const float*)d_in[1];
    const long long* cid = (const long long*)d_in[2];
    int B = in_sizes[0] / DIM;    // 256
    int N = in_sizes[2];          // 500000
    int CH = ((N + NCHUNK * RPS - 1) / (NCHUNK * RPS)) * RPS;  // chunk rows, mult of 128

    dim3 g1(B / 16, NCHUNK);
    knn_score_topk<<<g1, 256, 0, stream>>>(qe, corpus,
                                           (unsigned long long*)d_ws, N, CH);
    knn_merge<<<B, 256, 0, stream>>>((const unsigned long long*)d_ws,
                                     corpus, cid, d_out, B);
}